// GlobalContrastiveLoss_16501264351802
// MI455X (gfx1250) — compile-verified
//
#include <hip/hip_runtime.h>
#include <hip/hip_bf16.h>
#include <math.h>

typedef __attribute__((ext_vector_type(2))) float v2f;
typedef __attribute__((ext_vector_type(8))) float v8f;
typedef __attribute__((__vector_size__(16))) int v4i_t;
typedef __attribute__((address_space(1))) v4i_t* gv4i_p;   // global (AS1) int32x4*
typedef __attribute__((address_space(3))) v4i_t* lv4i_p;   // LDS (AS3) int32x4*

#define C_DIM 4096
#define D_DIM 1024
#define BM 128
#define BN 128
#define KC 32
#define LDSS 36            // padded row stride (floats): 144B = 9x16B, keeps b128 alignment
#define MARGIN_F 1.0f
#define EPS_F 1e-10f

// ---- CDNA5 async global->LDS (DMA, tracked by ASYNCcnt; no VGPR round-trip) ----
__device__ __forceinline__ void async_ld16(const float* g, float* l) {
    __builtin_amdgcn_global_load_async_to_lds_b128(
        (gv4i_p)(v4i_t*)g, (lv4i_p)(v4i_t*)l, /*imm offset*/0, /*cpol*/0);
}
__device__ __forceinline__ void wait_async0() {
    asm volatile("s_wait_asynccnt 0x0" ::: "memory");
}

// ---------------- kernel 1: build mask_i from annotations ----------------
__global__ void gcl_prep_masks(const long long* __restrict__ ann, int nAnn,
                               int* __restrict__ maskI) {
    for (int i = threadIdx.x; i < C_DIM; i += blockDim.x) maskI[i] = 0;
    __syncthreads();
    for (int i = threadIdx.x; i < nAnn; i += blockDim.x) {
        long long t = ann[(size_t)i * 5 + 4];
        if (t != -1) maskI[(int)t] = 1;   // all racing writers store 1 -> deterministic
    }
}

// ------------- kernel 2: row norms f2/p2 and mask_j = rowsum(P)!=0 -------------
__global__ __launch_bounds__(256)
void gcl_row_stats(const float* __restrict__ F, const float* __restrict__ P,
                   float* __restrict__ f2, float* __restrict__ p2,
                   int* __restrict__ maskJ) {
    const int row = blockIdx.x;
    const float* fr = F + (size_t)row * D_DIM;
    const float* pr = P + (size_t)row * D_DIM;
    float fs = 0.f, ps2 = 0.f, ps = 0.f;
    for (int c = threadIdx.x; c < D_DIM; c += blockDim.x) {
        float fv = fr[c], pv = pr[c];
        fs  = fmaf(fv, fv, fs);
        ps2 = fmaf(pv, pv, ps2);
        ps += pv;
    }
    __shared__ float s0[256], s1[256], s2[256];
    s0[threadIdx.x] = fs; s1[threadIdx.x] = ps2; s2[threadIdx.x] = ps;
    __syncthreads();
    for (int off = 128; off > 0; off >>= 1) {
        if ((int)threadIdx.x < off) {
            s0[threadIdx.x] += s0[threadIdx.x + off];
            s1[threadIdx.x] += s1[threadIdx.x + off];
            s2[threadIdx.x] += s2[threadIdx.x + off];
        }
        __syncthreads();
    }
    if (threadIdx.x == 0) {
        f2[row] = s0[0];
        p2[row] = s1[0];
        maskJ[row] = (s2[0] != 0.f) ? 1 : 0;
    }
}

// ------------- kernel 3: fused fp32-WMMA distance GEMM + loss reduction -------------
__global__ __launch_bounds__(256)
void gcl_fused_loss(const float* __restrict__ F, const float* __restrict__ P,
                    const float* __restrict__ f2, const float* __restrict__ p2,
                    const int* __restrict__ maskI, const int* __restrict__ maskJ,
                    float* __restrict__ partials) {
    __shared__ float As[2][BM * LDSS];   // double-buffered panels
    __shared__ float Bs[2][BN * LDSS];

    const int tid    = threadIdx.x;
    const int wave   = tid >> 5;          // 0..7
    const int lane   = tid & 31;
    const int laneLo = lane & 15;
    const int laneHi = lane >> 4;         // 0 or 1
    const int iBase  = blockIdx.y * BM;
    const int jBase  = blockIdx.x * BN;

    // panel-staging slots owned by this thread: slot = s*256+tid -> (row, float4-col)
    int srow[4], sc4[4];
    #pragma unroll
    for (int s = 0; s < 4; ++s) {
        const int slot = s * 256 + tid;   // 0..1023
        srow[s] = slot >> 3;              // 0..127
        sc4[s]  = slot & 7;               // 0..7
    }

    v8f zinit = {};
    v8f acc[8];
    #pragma unroll
    for (int j = 0; j < 8; ++j) acc[j] = zinit;

    // prologue: async-DMA panel 0 into buffer 0
    #pragma unroll
    for (int s = 0; s < 4; ++s) {
        async_ld16(F + (size_t)(iBase + srow[s]) * D_DIM + sc4[s] * 4,
                   &As[0][srow[s] * LDSS + sc4[s] * 4]);
        async_ld16(P + (size_t)(jBase + srow[s]) * D_DIM + sc4[s] * 4,
                   &Bs[0][srow[s] * LDSS + sc4[s] * 4]);
    }
    wait_async0();
    __syncthreads();

    int buf = 0;
    for (int k0 = 0; k0 < D_DIM; k0 += KC, buf ^= 1) {
        // issue async DMA for the NEXT panel into the other buffer (overlaps compute)
        const int kn = k0 + KC;
        if (kn < D_DIM) {
            #pragma unroll
            for (int s = 0; s < 4; ++s) {
                async_ld16(F + (size_t)(iBase + srow[s]) * D_DIM + kn + sc4[s] * 4,
                           &As[buf ^ 1][srow[s] * LDSS + sc4[s] * 4]);
                async_ld16(P + (size_t)(jBase + srow[s]) * D_DIM + kn + sc4[s] * 4,
                           &Bs[buf ^ 1][srow[s] * LDSS + sc4[s] * 4]);
            }
        }

        // compute on current buffer
        // A fragment: lane holds A[laneLo][kk + 2*laneHi + {0,1}]  (ISA 16x4 f32 layout)
        const float* aRow = &As[buf][(wave * 16 + laneLo) * LDSS];
        const float* bBase = &Bs[buf][laneLo * LDSS];
        #pragma unroll
        for (int kk = 0; kk < KC; kk += 4) {
            const v2f a = *reinterpret_cast<const v2f*>(&aRow[kk + 2 * laneHi]);
            v2f bfr[8];
            #pragma unroll
            for (int j = 0; j < 8; ++j)
                bfr[j] = *reinterpret_cast<const v2f*>(&bBase[j * 16 * LDSS + kk + 2 * laneHi]);
            #pragma unroll
            for (int j = 0; j < 8; ++j)
                acc[j] = __builtin_amdgcn_wmma_f32_16x16x4_f32(
                    false, a, false, bfr[j], (short)0, acc[j], false, false);
        }

        // drain this iteration's async DMAs, then rendezvous before buffer swap
        wait_async0();
        __syncthreads();
    }

    // ---- epilogue: map accumulators to (i,j), apply masks, msd, margin terms ----
    float intra = 0.f, inter = 0.f;
    const int gi0 = iBase + wave * 16 + 8 * laneHi;   // M = r + 8*laneHi
    int   miv[8];
    float f2v[8];
    #pragma unroll
    for (int r = 0; r < 8; ++r) { miv[r] = maskI[gi0 + r]; f2v[r] = f2[gi0 + r]; }

    #pragma unroll
    for (int j = 0; j < 8; ++j) {
        const int gj = jBase + j * 16 + laneLo;       // N = laneLo
        const int mj = maskJ[gj];
        const float p2v = p2[gj];
        #pragma unroll
        for (int r = 0; r < 8; ++r) {
            if (miv[r] && mj) {
                const int gi = gi0 + r;
                float msd = f2v[r] + p2v - 2.0f * acc[j][r];
                msd = fmaxf(msd, 0.f) * (1.0f / (float)D_DIM);
                if (gi == gj) {
                    intra += msd;
                } else {
                    const float sdist = sqrtf(fmaxf(msd, 1e-12f));
                    const float e  = MARGIN_F - sdist;
                    const float em = fmaxf(e, 0.f);
                    inter += (e * e) * (em * em);     // /MARGIN^2 == 1
                }
            }
        }
    }

    // ---- deterministic block reduction into per-block partials ----
    __syncthreads();
    float* red0 = &As[0][0];
    float* red1 = &As[1][0];
    red0[tid] = intra; red1[tid] = inter;
    __syncthreads();
    for (int off = 128; off > 0; off >>= 1) {
        if (tid < off) { red0[tid] += red0[tid + off]; red1[tid] += red1[tid + off]; }
        __syncthreads();
    }
    if (tid == 0) {
        const int bid = blockIdx.y * gridDim.x + blockIdx.x;
        partials[2 * bid]     = red0[0];
        partials[2 * bid + 1] = red1[0];
    }
}

// ------------- kernel 4: deterministic finalize -------------
__global__ __launch_bounds__(256)
void gcl_finalize(const float* __restrict__ partials, int nPart,
                  const int* __restrict__ maskI, const int* __restrict__ maskJ,
                  float* __restrict__ out) {
    const int tid = threadIdx.x;
    float ai = 0.f, an = 0.f, cIn = 0.f, cI = 0.f, cJ = 0.f;
    for (int i = tid; i < nPart; i += 256) {
        ai += partials[2 * i];
        an += partials[2 * i + 1];
    }
    for (int i = tid; i < C_DIM; i += 256) {
        const float mi = (float)maskI[i], mj = (float)maskJ[i];
        cI += mi; cJ += mj; cIn += mi * mj;
    }
    __shared__ float s0[256], s1[256], s2[256], s3[256], s4[256];
    s0[tid] = ai; s1[tid] = an; s2[tid] = cIn; s3[tid] = cI; s4[tid] = cJ;
    __syncthreads();
    for (int off = 128; off > 0; off >>= 1) {
        if (tid < off) {
            s0[tid] += s0[tid + off]; s1[tid] += s1[tid + off];
            s2[tid] += s2[tid + off]; s3[tid] += s3[tid + off];
            s4[tid] += s4[tid + off];
        }
        __syncthreads();
    }
    if (tid == 0) {
        const float intraCnt = s2[0];
        const float interCnt = s3[0] * s4[0] - s2[0];
        out[0] = s0[0] / (intraCnt + EPS_F);
        out[1] = s1[0] / (interCnt + EPS_F);
    }
}

extern "C" void kernel_launch(void* const* d_in, const int* in_sizes, int n_in,
                              void* d_out, int out_size, void* d_ws, size_t ws_size,
                              hipStream_t stream) {
    const float*      F   = (const float*)d_in[0];      // cls_features (4096,1024)
    const float*      P   = (const float*)d_in[1];      // prototypes   (4096,1024)
    const long long*  ann = (const long long*)d_in[2];  // annotations  (8,32,5) int64
    // d_in[3] = steps (unused by the math)
    float* out = (float*)d_out;                         // [intra_loss, inter_loss]

    char* ws = (char*)d_ws;
    float* f2     = (float*)ws;                 ws += C_DIM * sizeof(float);
    float* p2     = (float*)ws;                 ws += C_DIM * sizeof(float);
    int*   maskI  = (int*)ws;                   ws += C_DIM * sizeof(int);
    int*   maskJ  = (int*)ws;                   ws += C_DIM * sizeof(int);
    float* parts  = (float*)ws;                 // 2 * (32*32) floats

    const int nAnn = in_sizes[2] / 5;           // 256

    gcl_prep_masks<<<1, 256, 0, stream>>>(ann, nAnn, maskI);
    gcl_row_stats<<<C_DIM, 256, 0, stream>>>(F, P, f2, p2, maskJ);

    dim3 grid(C_DIM / BN, C_DIM / BM);          // 32 x 32 blocks
    gcl_fused_loss<<<grid, 256, 0, stream>>>(F, P, f2, p2, maskI, maskJ, parts);

    gcl_finalize<<<1, 256, 0, stream>>>(parts, (int)(grid.x * grid.y), maskI, maskJ, out);
}